// JustinArm07Net_71622874628546
// MI455X (gfx1250) — compile-verified
//
#include <hip/hip_runtime.h>
#include <stdint.h>

// Forward kinematics for 7-DOF arm: out[b] = 8 prefix frames of chained 4x4 affine transforms.
// Store-bandwidth-bound (1.02 GB out @ 23.3 TB/s ~= 46us floor). Compute exploits sparsity of
// the local transforms (~20 FMA/step). Output staged through XOR-swizzled wave-private LDS and
// drained with gfx1250 async LDS->global B128 stores (512B fully-coalesced per instruction).

#define WAVES_PER_BLOCK 4
#define BLOCK_THREADS   (WAVES_PER_BLOCK * 32)

struct F3 { float x, y, z; };

__device__ __forceinline__ F3 f3(float a, float b, float c) { return {a, b, c}; }
__device__ __forceinline__ F3 neg(F3 a) { return {-a.x, -a.y, -a.z}; }
__device__ __forceinline__ F3 mul(float s, F3 a) { return {s * a.x, s * a.y, s * a.z}; }
__device__ __forceinline__ F3 fma3(float s, F3 a, F3 acc) {
    return { fmaf(s, a.x, acc.x), fmaf(s, a.y, acc.y), fmaf(s, a.z, acc.z) };
}

__global__ __launch_bounds__(BLOCK_THREADS) void fk7_kernel(const float* __restrict__ q,
                                                            float* __restrict__ out,
                                                            int n) {
    // 4 waves * 32 elems * 32 float4 = 64 KB; each wave owns a private 16 KB slab.
    __shared__ __align__(16) float4 lds[WAVES_PER_BLOCK * 32 * 32];

    const int lane = threadIdx.x & 31;
    const int wib  = threadIdx.x >> 5;
    const long long waveId = (long long)blockIdx.x * WAVES_PER_BLOCK + wib;
    const long long base   = waveId * 32;          // first element of this wave's tile
    const long long elem   = base + lane;
    const long long eIdx   = (elem < n) ? elem : (long long)(n - 1);

    // ---- load joint angles, 7 sincos ----
    const float* qp = q + eIdx * 7;
    float s[7], c[7];
#pragma unroll
    for (int i = 0; i < 7; ++i) __sincosf(qp[i], &s[i], &c[i]);

    float4* slab = lds + wib * (32 * 32);

    // Frame columns (rows 0..2 only; row 3 is always (0,0,0,1)).
    F3 X = f3( c[0], s[0], 0.f);
    F3 Y = f3(-s[0], c[0], 0.f);
    F3 Z = f3(0.f, 0.f, 1.f);
    F3 P = f3(0.f, 0.f, 0.f);

    // LDS layout: chunk m (= frame*4 + row) of lane t lives at slot m*32 + (t ^ m).
    // -> DS write side: fixed m across lanes = contiguous 512B (permuted), conflict-minimal.
    // -> async read side (transposed) also hits the 2-cycle B128 bank minimum.
    auto storeFrame = [&](int f) {
        const int m = f * 4;
        slab[(m + 0) * 32 + (lane ^ (m + 0))] = make_float4(X.x, Y.x, Z.x, P.x);
        slab[(m + 1) * 32 + (lane ^ (m + 1))] = make_float4(X.y, Y.y, Z.y, P.y);
        slab[(m + 2) * 32 + (lane ^ (m + 2))] = make_float4(X.z, Y.z, Z.z, P.z);
        slab[(m + 3) * 32 + (lane ^ (m + 3))] = make_float4(0.f, 0.f, 0.f, 1.f);
    };

    // Sparse chain steps (F' = F * M with M's fixed sparsity):
    // pattern A (m1,m3,m5,m6): X'=c*X+s*Z; Y'=-s*X+c*Z; Z'=-Y
    auto stepA = [&](float cc, float ss) {
        F3 Xn = fma3(ss, Z, mul(cc, X));
        F3 Yn = fma3(cc, Z, mul(-ss, X));
        F3 Zn = neg(Y);
        X = Xn; Y = Yn; Z = Zn;
    };
    // pattern B (m2,m4): X'=c*X-s*Z; Y'=-s*X-c*Z; Z'=Y; P'+=d*Y
    auto stepB = [&](float cc, float ss, float d) {
        F3 Xn = fma3(-ss, Z, mul(cc, X));
        F3 Yn = fma3(-cc, Z, mul(-ss, X));
        F3 Zn = Y;
        P = fma3(d, Y, P);
        X = Xn; Y = Yn; Z = Zn;
    };

    storeFrame(0);
    stepA(c[1], s[1]);                 storeFrame(1);
    // m2: angle q2 - pi/2 -> cos = sin(q2), sin = -cos(q2)
    stepB(s[2], -c[2], 0.40f);         storeFrame(2);
    stepA(c[3], s[3]);                 storeFrame(3);
    // m4 == pattern B with (c,s) -> (-c4, -s4), d = 0.39
    stepB(-c[4], -s[4], 0.39f);        storeFrame(4);
    // m5: angle q5 + pi/2 -> cos = -sin(q5), sin = cos(q5)
    stepA(-s[5], c[5]);                storeFrame(5);
    // m6: angle q6 - pi/2 -> cos = sin(q6), sin = -cos(q6)
    stepA(s[6], -c[6]);                storeFrame(6);
    // m7: X'=-X; Y'=Z; Z'=Y; P'+=0.118*Y
    {
        F3 Xn = neg(X); F3 Yn = Z; F3 Zn = Y;
        P = fma3(0.118f, Y, P);
        X = Xn; Y = Yn; Z = Zn;
    }
    storeFrame(7);

    // All LDS traffic is wave-private: wait for this wave's DS writes only (no barrier).
    asm volatile("s_wait_dscnt 0x0" ::: "memory");

    // Drain the 16 KB tile with async LDS->global stores: instruction i stores element
    // (base+i)'s entire 512B block; 32 lanes x 16B = 512B contiguous global bytes.
    float4* tile = (float4*)out + base * 32;   // 32 float4 per element
    long long rem = (long long)n - base;
    int valid = rem < 32 ? (rem < 0 ? 0 : (int)rem) : 32;
#pragma unroll 1
    for (int i = 0; i < valid; ++i) {
        uint64_t gaddr = (uint64_t)(uintptr_t)(tile + (i * 32 + lane));
        uint32_t laddr = (uint32_t)(uintptr_t)(slab + (lane * 32 + (i ^ lane)));
        asm volatile("global_store_async_from_lds_b128 %0, %1, off"
                     :: "v"(gaddr), "v"(laddr)
                     : "memory");
    }
    asm volatile("s_wait_asynccnt 0x0" ::: "memory");
}

extern "C" void kernel_launch(void* const* d_in, const int* in_sizes, int n_in,
                              void* d_out, int out_size, void* d_ws, size_t ws_size,
                              hipStream_t stream) {
    (void)n_in; (void)out_size; (void)d_ws; (void)ws_size;
    const float* x = (const float*)d_in[0];
    float* out = (float*)d_out;
    const int n = in_sizes[0] / 7;                       // 2,000,000 elements
    const long long waves  = ((long long)n + 31) / 32;
    const int blocks = (int)((waves + WAVES_PER_BLOCK - 1) / WAVES_PER_BLOCK);
    fk7_kernel<<<blocks, BLOCK_THREADS, 0, stream>>>(x, out, n);
}